// OhemCELoss_66718021976386
// MI455X (gfx1250) — compile-verified
//
#include <hip/hip_runtime.h>
#include <hip/hip_bf16.h>
#include <math.h>

#define GLOBAL_AS __attribute__((address_space(1)))
#define LDS_AS    __attribute__((address_space(3)))

#if defined(__has_builtin)
#  if __has_builtin(__builtin_amdgcn_global_load_async_to_lds_b32)
#    define HAVE_ASYNC_LDS 1
#  endif
#endif
#ifndef HAVE_ASYNC_LDS
#  define HAVE_ASYNC_LDS 0
#endif

static constexpr int   kN = 8, kC = 19, kH = 512, kW = 1024;
static constexpr int   kHW    = kH * kW;            // 524288 = 2^19
static constexpr int   kNPIX  = kN * kHW;           // 4194304
static constexpr int   kNMIN  = kNPIX / 16;         // 262144
static constexpr float kTHRESH = 0.35667494393873245f;   // -log(0.7)
static constexpr int   kNBINS = 1024;
static constexpr float kBINSCALE = 64.0f;           // bins cover [0,16)
static constexpr float kFIX   = 1048576.0f;         // 2^20 fixed-point scale
static constexpr int   kBLOCK = 256;                // 8 wave32s
static constexpr int   kNTILES = kNPIX / kBLOCK;    // 16384
static constexpr int   kGRID  = 1024;

// Workspace layout (bytes):
//   [0]                      u64 sumFixed      (hard-pixel loss sum, 2^20 fixed point)
//   [8]                      u32 countHard
//   [16]                     u32 histCnt[1024]
//   [16 + 4*kNBINS]          u64 histSum[1024] (fixed point)
static constexpr int WS_BYTES = 16 + 4 * kNBINS + 8 * kNBINS;   // 12304
static constexpr int WS_WORDS = WS_BYTES / 4;                   // 3076

template <int N>
__device__ __forceinline__ void wait_async_le() {
#if defined(__has_builtin) && __has_builtin(__builtin_amdgcn_s_wait_asynccnt)
  __builtin_amdgcn_s_wait_asynccnt(N);
#elif HAVE_ASYNC_LDS
  asm volatile("s_wait_asynccnt %0" ::"i"(N) : "memory");
#endif
}

__global__ __launch_bounds__(kBLOCK)
void ohem_main(const float* __restrict__ logits, const int* __restrict__ labels,
               unsigned long long* __restrict__ sumFixed, unsigned* __restrict__ countHard,
               unsigned* __restrict__ gHistCnt, unsigned long long* __restrict__ gHistSum) {
  __shared__ unsigned            sCnt[kNBINS];
  __shared__ unsigned long long  sSum[kNBINS];
  __shared__ unsigned long long  sHardSum;
  __shared__ unsigned            sHardCnt;
#if HAVE_ASYNC_LDS
  __shared__ float stage[2][kC][kBLOCK];   // 38,912 B double-buffered channel tiles
#endif

  const int tid = threadIdx.x;
  for (int b = tid; b < kNBINS; b += kBLOCK) { sCnt[b] = 0u; sSum[b] = 0ull; }
  if (tid == 0) { sHardSum = 0ull; sHardCnt = 0u; }
  __syncthreads();

  // Per-pixel accumulate into block-local LDS stats (integer => deterministic).
  auto accumulate = [&](float loss) {
    loss = fmaxf(loss, 0.0f);
    unsigned long long fx = (unsigned long long)(loss * kFIX + 0.5f);
    if (loss > kTHRESH) {
      atomicAdd(&sHardCnt, 1u);
      atomicAdd(&sHardSum, fx);
    }
    int b = (int)(loss * kBINSCALE);
    if (b > kNBINS - 1) b = kNBINS - 1;
    atomicAdd(&sCnt[b], 1u);
    atomicAdd(&sSum[b], fx);
  };

#if HAVE_ASYNC_LDS
  // Issue 19 async DMA channel loads for this thread's pixel of `tile` into buffer `sel`.
  // Each thread consumes only its own staged slots -> no block barrier needed,
  // only the per-wave ASYNCcnt wait.
  auto issue = [&](int tile, int sel) {
    int p   = tile * kBLOCK + tid;            // pixel id, < 2^22
    int n   = p >> 19;                        // p / kHW
    int rem = p & (kHW - 1);
    const float* g = logits + ((long)n * kC) * kHW + rem;
    float* l = &stage[sel][0][tid];
#pragma unroll
    for (int c = 0; c < kC; ++c) {
      __builtin_amdgcn_global_load_async_to_lds_b32(
          (GLOBAL_AS int*)(g + (long)c * kHW),
          (LDS_AS int*)(l + c * kBLOCK),
          0, 0);
    }
  };

  auto compute_staged = [&](int tile, int sel) {
    int p   = tile * kBLOCK + tid;
    int lab = labels[p];
    float m = -INFINITY, xl = 0.0f;
#pragma unroll
    for (int c = 0; c < kC; ++c) {            // pass 1: max + label gather (no reg array)
      float v = stage[sel][c][tid];
      m  = fmaxf(m, v);
      xl = (c == lab) ? v : xl;
    }
    float s = 0.0f;
#pragma unroll
    for (int c = 0; c < kC; ++c) {            // pass 2: sum exp (CSE'd from LDS pass 1)
      float v = stage[sel][c][tid];
      s += expf(v - m);
    }
    accumulate(m + logf(s) - xl);
  };

  int sel = 0;
  int t0  = blockIdx.x;
  if (t0 < kNTILES) issue(t0, 0);
  for (int t = t0; t < kNTILES; t += gridDim.x) {
    int tn = t + gridDim.x;
    if (tn < kNTILES) {
      issue(tn, sel ^ 1);       // prefetch next tile into the other buffer
      wait_async_le<kC>();      // first 19 (current buffer) retired; next 19 in flight
    } else {
      wait_async_le<0>();
    }
    compute_staged(t, sel);
    sel ^= 1;
  }
#else
  // Fallback: direct coalesced global loads, values held in registers.
  for (int t = blockIdx.x; t < kNTILES; t += gridDim.x) {
    int p   = t * kBLOCK + tid;
    int n   = p >> 19;
    int rem = p & (kHW - 1);
    const float* g = logits + ((long)n * kC) * kHW + rem;
    int lab = labels[p];
    float x[kC];
#pragma unroll
    for (int c = 0; c < kC; ++c) x[c] = g[(long)c * kHW];
    float m = -INFINITY, xl = 0.0f;
#pragma unroll
    for (int c = 0; c < kC; ++c) { m = fmaxf(m, x[c]); xl = (c == lab) ? x[c] : xl; }
    float s = 0.0f;
#pragma unroll
    for (int c = 0; c < kC; ++c) s += expf(x[c] - m);
    accumulate(m + logf(s) - xl);
  }
#endif

  __syncthreads();
  if (tid == 0 && sHardCnt) {
    atomicAdd(countHard, sHardCnt);
    atomicAdd(sumFixed, sHardSum);
  }
  for (int b = tid; b < kNBINS; b += kBLOCK) {
    unsigned c = sCnt[b];
    if (c) {
      atomicAdd(&gHistCnt[b], c);
      atomicAdd(&gHistSum[b], sSum[b]);
    }
  }
}

__global__ void ohem_zero_ws(unsigned* __restrict__ ws) {
  int i = blockIdx.x * blockDim.x + threadIdx.x;
  if (i < WS_WORDS) ws[i] = 0u;
}

__global__ void ohem_finalize(const unsigned long long* __restrict__ sumFixed,
                              const unsigned* __restrict__ countHard,
                              const unsigned* __restrict__ gHistCnt,
                              const unsigned long long* __restrict__ gHistSum,
                              float* __restrict__ out) {
  if (threadIdx.x != 0 || blockIdx.x != 0) return;
  unsigned cnt = *countHard;
  double r;
  if (cnt >= (unsigned)kNMIN) {
    r = ((double)*sumFixed / (double)kFIX) / (double)cnt;
  } else {
    // Approximate top-k mean via histogram scan from the top bin.
    long need = kNMIN;
    double acc = 0.0;
    for (int b = kNBINS - 1; b >= 0 && need > 0; --b) {
      unsigned c = gHistCnt[b];
      if (!c) continue;
      double s = (double)gHistSum[b] / (double)kFIX;
      if ((long)c <= need) { acc += s; need -= (long)c; }
      else                 { acc += s * ((double)need / (double)c); need = 0; }
    }
    r = acc / (double)kNMIN;
  }
  *out = (float)r;
}

extern "C" void kernel_launch(void* const* d_in, const int* in_sizes, int n_in,
                              void* d_out, int out_size, void* d_ws, size_t ws_size,
                              hipStream_t stream) {
  (void)in_sizes; (void)n_in; (void)out_size; (void)ws_size;
  const float* logits = (const float*)d_in[0];
  const int*   labels = (const int*)d_in[1];

  unsigned char* ws = (unsigned char*)d_ws;
  unsigned long long* sumFixed  = (unsigned long long*)(ws + 0);
  unsigned*           countHard = (unsigned*)(ws + 8);
  unsigned*           hCnt      = (unsigned*)(ws + 16);
  unsigned long long* hSum      = (unsigned long long*)(ws + 16 + 4 * kNBINS);

  ohem_zero_ws<<<(WS_WORDS + 255) / 256, 256, 0, stream>>>((unsigned*)ws);
  ohem_main<<<kGRID, kBLOCK, 0, stream>>>(logits, labels, sumFixed, countHard, hCnt, hSum);
  ohem_finalize<<<1, 64, 0, stream>>>(sumFixed, countHard, hCnt, hSum, (float*)d_out);
}